// MildAugmentStage_28484223108031
// MI455X (gfx1250) — compile-verified
//
#include <hip/hip_runtime.h>
#include <cstdint>
#include <cstddef>

// ---------------------------------------------------------------------------
// MI455X (gfx1250) implementation of the MildAugment pipeline.
//
// All resampling passes (cubic crop-resize H, W; Gaussian blur H, W) are
// banded-matrix x image GEMMs executed with V_WMMA_F32_16X16X4_F32 (fp32
// in/out keeps reference precision; the pipeline is HBM-bound at ~1.1 GB of
// traffic, so lower-precision WMMA buys nothing). LDS tiles are staged with
// GLOBAL_LOAD_ASYNC_TO_LDS_B128 (ASYNCcnt) and s_wait_asynccnt.
//
// Workspace layout (bytes), total = 352,403,456 (~336 MB):
//   P   params        @ 0         256 * 32 f32
//   SH  resizeH starts@ 32768     256*14 i32
//   SW  resizeW starts@ 49152
//   SB  blur starts   @ 65536
//   WH  resizeH tiles @ 81920     256*14*16*32 f32 (7.34 MB)
//   WW  resizeW tiles @ 7421952
//   WB  blur tiles    @ 14761984
//   B1  intermediate  @ 22102016  256*3*224*256 f32 (176.2 MB)
//   B2  xc / xj       @ 198262784 256*3*224*224 f32 (154.1 MB)
// ---------------------------------------------------------------------------

typedef float v2f __attribute__((ext_vector_type(2)));
typedef float v8f __attribute__((ext_vector_type(8)));

#define BATCH   256
#define HW_IN   256
#define RES     224
#define PIX     (RES * RES)          // 50176
#define NTILE   (RES / 16)           // 14

static const size_t OFF_SH = 32768;
static const size_t OFF_SW = 49152;
static const size_t OFF_SB = 65536;
static const size_t OFF_WH = 81920;
static const size_t OFF_WW = 7421952;
static const size_t OFF_WB = 14761984;
static const size_t OFF_B1 = 22102016;
static const size_t OFF_B2 = 198262784;

// --------------------------- helpers ---------------------------------------

__device__ __forceinline__ float clip01(float x) {
    return fminf(fmaxf(x, 0.f), 1.f);
}

// Deterministic counter-based uniform in [0,1) (SplitMix64 hash); stands in
// for the reference's threefry streams.
__device__ __forceinline__ float urand(unsigned stream, unsigned idx) {
    unsigned long long z =
        ((unsigned long long)stream * 0x9E3779B97F4A7C15ULL) ^
        ((unsigned long long)idx * 0xD1B54A32D192ED03ULL + 0x632BE59BD9B4E019ULL);
    z ^= z >> 30; z *= 0xBF58476D1CE4E5B9ULL;
    z ^= z >> 27; z *= 0x94D049BB133111EBULL;
    z ^= z >> 31;
    return (float)(unsigned)(z >> 40) * (1.0f / 16777216.0f);
}

// Keys cubic, a = -0.5 (matches jax.image 'cubic').
__device__ __forceinline__ float cubic_keys(float t) {
    t = fabsf(t);
    float t2 = t * t, t3 = t2 * t;
    if (t <= 1.f) return 1.5f * t3 - 2.5f * t2 + 1.f;
    if (t <  2.f) return -0.5f * t3 + 2.5f * t2 - 4.f * t + 2.f;
    return 0.f;
}

// Issue one async global->LDS 128-bit copy per lane (32 lanes x 16B = 512B).
// Tracked with ASYNCcnt; consumer must s_wait_asynccnt + barrier.
__device__ __forceinline__ void async_copy_b128(unsigned lds_addr,
                                                unsigned long long gaddr) {
    asm volatile("global_load_async_to_lds_b128 %0, %1, off"
                 :: "v"(lds_addr), "v"(gaddr) : "memory");
}
__device__ __forceinline__ void async_wait0() {
    asm volatile("s_wait_asynccnt 0x0" ::: "memory");
}
__device__ __forceinline__ unsigned lds_offset(const void* p) {
    // Flat shared-aperture addresses carry the LDS byte offset in the low
    // 32 bits (ISA 10.2).
    return (unsigned)(unsigned long long)(uintptr_t)p;
}

// --------------------------- per-batch params -------------------------------

__global__ void params_kernel(float* __restrict__ P) {
    int b = threadIdx.x;
    if (b >= BATCH) return;
    float* p = P + b * 32;

    const float Hf = (float)HW_IN, Wf = (float)HW_IN, INf = (float)RES;
    float area  = Hf * Wf * (0.8f + 0.2f * urand(0, b));
    float lr = logf(3.f / 4.f), hr = logf(4.f / 3.f);
    float ratio = expf(lr + (hr - lr) * urand(1, b));
    float cw = fminf(fmaxf(rintf(sqrtf(area * ratio)), 1.f), Wf);
    float ch = fminf(fmaxf(rintf(sqrtf(area / ratio)), 1.f), Hf);
    float i0 = floorf(urand(2, b) * (Hf - ch + 1.f));
    float j0 = floorf(urand(3, b) * (Wf - cw + 1.f));

    p[0] = INf / ch;                        // sy
    p[1] = INf / cw;                        // sx
    p[2] = i0;
    p[3] = j0;
    p[4] = 0.9f + 0.2f * urand(4, b);       // brightness
    p[5] = 0.9f + 0.2f * urand(5, b);       // contrast
    p[6] = 0.9f + 0.2f * urand(6, b);       // saturation
    p[7] = -0.01f + 0.02f * urand(7, b);    // hue
    p[8] = (urand(8, b) < 0.8f) ? 1.f : 0.f; // apply color jitter
    float sigma = 0.1f + 1.9f * urand(9, b);
    p[9]  = (urand(10, b) < 0.5f) ? 1.f : 0.f; // apply blur
    p[10] = 0.f;                            // mean gray (filled later)
    p[11] = sigma;

    float ks = 0.f, kern[15];
    for (int k = 0; k < 15; ++k) {
        float t = (float)(k - 7) / sigma;
        kern[k] = expf(-0.5f * t * t);
        ks += kern[k];
    }
    float inv = 1.f / ks;
    for (int k = 0; k < 15; ++k) p[16 + k] = kern[k] * inv;
}

// ----------------------- compact banded weight tiles ------------------------
// One block per (b, output-row-tile t). Threads 0-15: resize-H rows,
// 16-31: resize-W rows, 32-47: blur rows (reflect-folded band).

__global__ void build_weights(const float* __restrict__ P,
                              float* __restrict__ WH, float* __restrict__ WW,
                              float* __restrict__ WB,
                              int* __restrict__ SH, int* __restrict__ SW,
                              int* __restrict__ SB) {
    int b = blockIdx.x, t = blockIdx.y, tid = threadIdx.x;
    const float* p = P + b * 32;

    if (tid < 32) {                       // cubic antialias resize weights
        int axis = tid >> 4;              // 0 = H, 1 = W
        int m    = tid & 15;
        float sc = p[axis];               // sy or sx
        float o0 = p[2 + axis];           // i0 or j0
        float s  = fminf(sc, 1.f);        // antialias kernel scale
        float pos0 = o0 + ((float)(t * 16) + 0.5f) / sc - 0.5f;
        int start = (int)floorf(pos0) - 4;
        if (start < 0) start = 0;
        if (start > HW_IN - 32) start = HW_IN - 32;
        start &= ~3;                      // 16B-aligned window for async b128
        float pos = o0 + ((float)(t * 16 + m) + 0.5f) / sc - 0.5f;
        float w[32], sum = 0.f;
        for (int j = 0; j < 32; ++j) {
            float v = cubic_keys(((float)(start + j) - pos) * s);
            w[j] = v; sum += v;
        }
        float inv = 1.f / fmaxf(sum, 1e-8f);
        float* d = (axis ? WW : WH) + (size_t)(b * NTILE + t) * 512 + m * 32;
        for (int j = 0; j < 32; ++j) d[j] = w[j] * inv;
        if (m == 0) (axis ? SW : SH)[b * NTILE + t] = start;
    } else if (tid < 48) {                // Gaussian band, reflect folded
        int m  = tid - 32;
        int oy = t * 16 + m;
        int start = t * 16 - 8;
        if (start < 0) start = 0;
        if (start > RES - 32) start = RES - 32;   // stays a multiple of 8
        float w[32];
        for (int j = 0; j < 32; ++j) w[j] = 0.f;
        for (int k = 0; k < 15; ++k) {
            int j = oy + k - 7;
            if (j < 0) j = -j;                       // 'reflect' pad
            if (j > RES - 1) j = 2 * (RES - 1) - j;
            w[j - start] += p[16 + k];
        }
        float* d = WB + (size_t)(b * NTILE + t) * 512 + m * 32;
        for (int j = 0; j < 32; ++j) d[j] = w[j];
        if (m == 0) SB[b * NTILE + t] = start;
    }
}

// --------------------- WMMA GEMM, weights on the left -----------------------
// out[t*16+m][col] = sum_K W[m][K] * in[start+K][col]   (resize-H / blur-H)
// 8 waves/block, each owns one 16-wide column tile; 8 K-steps of f32 16x16x4.
// All strides are template constants -> loads use immediate offsets.

template<int IN_RS, long long IN_IMG, int OUT_RS, long long OUT_IMG, int NTC>
__global__ void gemm_rows(const float* __restrict__ in, float* __restrict__ out,
                          const float* __restrict__ Wt,
                          const int* __restrict__ starts) {
    __shared__ __align__(16) float sW[512];       // 16x32 weight tile
    int tid  = threadIdx.x;
    int img  = blockIdx.x;                        // b*3 + c
    int b    = img / 3;
    int t    = blockIdx.y;
    int wave = tid >> 5, lane = tid & 31;

    // Async-stage the 2 KB weight tile into LDS (wave 0, 4 x B128).
    const float* wsrc = Wt + (size_t)(b * NTILE + t) * 512;
    if (wave == 0) {
        unsigned lds = lds_offset(&sW[0]) + (unsigned)lane * 16u;
        unsigned long long ga = (unsigned long long)(uintptr_t)wsrc + (unsigned long long)lane * 16ull;
        async_copy_b128(lds, ga);
        asm volatile("global_load_async_to_lds_b128 %0, %1, off offset:512"
                     :: "v"(lds), "v"(ga) : "memory");
        asm volatile("global_load_async_to_lds_b128 %0, %1, off offset:1024"
                     :: "v"(lds), "v"(ga) : "memory");
        asm volatile("global_load_async_to_lds_b128 %0, %1, off offset:1536"
                     :: "v"(lds), "v"(ga) : "memory");
    }
    async_wait0();
    __syncthreads();

    int ct  = blockIdx.z * 8 + wave;
    int ctc = (ct < NTC) ? ct : (NTC - 1);        // clamp: EXEC stays all-ones
    int m   = lane & 15;
    int hx  = lane >> 4;
    int start = starts[b * NTILE + t];

    // Hoisted bases; the unrolled K loop uses constant immediate offsets.
    const float* pb = in + (size_t)img * IN_IMG
                         + (size_t)(start + hx * 2) * IN_RS + ctc * 16 + m;
    const float* wp = &sW[m * 32 + hx * 2];
    __builtin_prefetch(pb + 16 * IN_RS, 0, 0);

    v8f acc = {};
#pragma unroll
    for (int kk = 0; kk < 8; ++kk) {
        v2f a = *(const v2f*)(wp + kk * 4);       // ds_load_b64
        v2f bb;
        bb.x = pb[kk * 4 * IN_RS];
        bb.y = pb[kk * 4 * IN_RS + IN_RS];
        acc = __builtin_amdgcn_wmma_f32_16x16x4_f32(
                  false, a, false, bb, (short)0, acc, false, false);
    }
    if (ct < NTC) {
        float* op = out + (size_t)img * OUT_IMG
                        + (size_t)(t * 16 + 8 * hx) * OUT_RS + ct * 16 + m;
#pragma unroll
        for (int r = 0; r < 8; ++r)
            op[r * OUT_RS] = acc[r];
    }
}

// --------------------- WMMA GEMM, weights on the right ----------------------
// out[row][t2*16+n] = sum_K in[row][start+K] * W[n][K]   (resize-W / blur-W)
// Per-wave 16x32 A tile staged via async B128 copies into LDS.

template<int IN_RS, long long IN_IMG, int OUT_RS, long long OUT_IMG, bool FUSE>
__global__ void gemm_cols(const float* __restrict__ in, float* __restrict__ out,
                          const float* __restrict__ Wt,
                          const int* __restrict__ starts,
                          const float* __restrict__ xj,
                          const float* __restrict__ P) {
    __shared__ __align__(16) float sW[512];
    __shared__ __align__(16) float sA[8][16][32];
    int tid  = threadIdx.x;
    int img  = blockIdx.x, b = img / 3;
    int t2   = blockIdx.y;
    int wave = tid >> 5, lane = tid & 31;

    const float* wsrc = Wt + (size_t)(b * NTILE + t2) * 512;
    if (wave == 0) {
        unsigned lds = lds_offset(&sW[0]) + (unsigned)lane * 16u;
        unsigned long long ga = (unsigned long long)(uintptr_t)wsrc + (unsigned long long)lane * 16ull;
        async_copy_b128(lds, ga);
        asm volatile("global_load_async_to_lds_b128 %0, %1, off offset:512"
                     :: "v"(lds), "v"(ga) : "memory");
        asm volatile("global_load_async_to_lds_b128 %0, %1, off offset:1024"
                     :: "v"(lds), "v"(ga) : "memory");
        asm volatile("global_load_async_to_lds_b128 %0, %1, off offset:1536"
                     :: "v"(lds), "v"(ga) : "memory");
    }

    int rt    = blockIdx.z * 8 + wave;
    int rtc   = (rt < NTILE) ? rt : (NTILE - 1);
    int start = starts[b * NTILE + t2];

    // Stage this wave's 16x32 A tile (rows rtc*16.., cols start..start+31):
    // lane -> (row = j*4 + lane/8, col = (lane&7)*4), 4 x B128 per wave.
    const float* gbase = in + (size_t)img * IN_IMG
                            + (size_t)(rtc * 16) * IN_RS + start;
    unsigned sAbase = lds_offset(&sA[wave][0][0]);
    int arow = lane >> 3;
    int acol = (lane & 7) * 4;
#pragma unroll
    for (int j = 0; j < 4; ++j) {
        int row = j * 4 + arow;
        unsigned long long ga =
            (unsigned long long)(uintptr_t)(gbase + (size_t)row * IN_RS + acol);
        unsigned lv = sAbase + (unsigned)(row * 32 + acol) * 4u;
        async_copy_b128(lv, ga);
    }
    async_wait0();
    __syncthreads();

    int m  = lane & 15;
    int hx = lane >> 4;
    const float* ap = &sA[wave][m][hx * 2];
    const float* wp = &sW[m * 32 + hx * 2];

    v8f acc = {};
#pragma unroll
    for (int kk = 0; kk < 8; ++kk) {
        v2f a  = *(const v2f*)(ap + kk * 4);
        v2f bb = *(const v2f*)(wp + kk * 4);
        acc = __builtin_amdgcn_wmma_f32_16x16x4_f32(
                  false, a, false, bb, (short)0, acc, false, false);
    }
    if (rt < NTILE) {
        float* op = out + (size_t)img * OUT_IMG
                        + (size_t)(rt * 16 + 8 * hx) * OUT_RS + t2 * 16 + m;
        if (!FUSE) {
#pragma unroll
            for (int r = 0; r < 8; ++r)
                op[r * OUT_RS] = acc[r];
        } else {                           // final blend: where(apply_bl, xb, xj)
            const float* xjp = xj + (size_t)img * OUT_IMG
                                  + (size_t)(rt * 16 + 8 * hx) * OUT_RS + t2 * 16 + m;
            float abl = P[b * 32 + 9];
#pragma unroll
            for (int r = 0; r < 8; ++r)
                op[r * OUT_RS] = (abl > 0.5f) ? acc[r] : xjp[r * OUT_RS];
        }
    }
}

// --------------------- per-image mean of grayscale --------------------------

__global__ void meangray_kernel(const float* __restrict__ xc, float* __restrict__ P) {
    __shared__ float red[256];
    int b = blockIdx.x, tid = threadIdx.x;
    const float* base = xc + (size_t)b * 3 * PIX;
    float bf = P[b * 32 + 4];
    float s = 0.f;
    for (int p = tid; p < PIX; p += 256) {
        float r  = clip01(base[p] * bf);
        float g  = clip01(base[p + PIX] * bf);
        float bl = clip01(base[p + 2 * PIX] * bf);
        s += 0.2989f * r + 0.587f * g + 0.114f * bl;
    }
    red[tid] = s;
    __syncthreads();
    for (int o = 128; o > 0; o >>= 1) {
        if (tid < o) red[tid] += red[tid + o];
        __syncthreads();
    }
    if (tid == 0) P[b * 32 + 10] = red[0] * (1.f / (float)PIX);
}

// --------------------- fused color jitter (in place) ------------------------

__global__ void jitter_kernel(float* __restrict__ img, const float* __restrict__ P) {
    int gid = blockIdx.x * blockDim.x + threadIdx.x;
    int b = gid / PIX, p = gid % PIX;
    const float* prm = P + b * 32;
    float bf = prm[4], cf = prm[5], sf = prm[6], hfac = prm[7];
    float acj = prm[8], mean = prm[10];

    float* base = img + (size_t)b * 3 * PIX;
    float r0 = base[p], g0 = base[p + PIX], b0 = base[p + 2 * PIX];

    float r  = clip01(r0 * bf);
    float g  = clip01(g0 * bf);
    float bl = clip01(b0 * bf);
    r  = clip01(cf * r  + (1.f - cf) * mean);
    g  = clip01(cf * g  + (1.f - cf) * mean);
    bl = clip01(cf * bl + (1.f - cf) * mean);
    float gray = 0.2989f * r + 0.587f * g + 0.114f * bl;
    r  = clip01(sf * r  + (1.f - sf) * gray);
    g  = clip01(sf * g  + (1.f - sf) * gray);
    bl = clip01(sf * bl + (1.f - sf) * gray);

    // rgb -> hsv
    float maxc = fmaxf(r, fmaxf(g, bl)), minc = fminf(r, fminf(g, bl));
    float v = maxc, d = maxc - minc;
    float sat = (maxc > 0.f) ? d / maxc : 0.f;
    float dn = (d > 0.f) ? d : 1.f;
    float rc = (maxc - r) / dn, gc = (maxc - g) / dn, bc = (maxc - bl) / dn;
    float h = (maxc == r) ? (bc - gc) : ((maxc == g) ? (2.f + rc - bc) : (4.f + gc - rc));
    h = (d > 0.f) ? (h * (1.f / 6.f)) : 0.f;
    h -= floorf(h);
    h += hfac;
    h -= floorf(h);

    // hsv -> rgb
    float h6 = h * 6.f;
    float fi = floorf(h6);
    float f  = h6 - fi;
    int i = ((int)fi) % 6;
    if (i < 0) i += 6;
    float pp = v * (1.f - sat), q = v * (1.f - sat * f), tt = v * (1.f - sat * (1.f - f));
    float rr, gg, bb;
    switch (i) {
        case 0: rr = v;  gg = tt; bb = pp; break;
        case 1: rr = q;  gg = v;  bb = pp; break;
        case 2: rr = pp; gg = v;  bb = tt; break;
        case 3: rr = pp; gg = q;  bb = v;  break;
        case 4: rr = tt; gg = pp; bb = v;  break;
        default: rr = v; gg = pp; bb = q;  break;
    }
    bool cj = acj > 0.5f;
    base[p]           = cj ? rr : r0;
    base[p + PIX]     = cj ? gg : g0;
    base[p + 2 * PIX] = cj ? bb : b0;
}

// ----------------------------- launcher -------------------------------------

extern "C" void kernel_launch(void* const* d_in, const int* in_sizes, int n_in,
                              void* d_out, int out_size, void* d_ws, size_t ws_size,
                              hipStream_t stream) {
    (void)in_sizes; (void)n_in; (void)out_size; (void)ws_size;
    const float* x = (const float*)d_in[0];
    char* ws = (char*)d_ws;
    float* P  = (float*)ws;
    int* SH   = (int*)(ws + OFF_SH);
    int* SWt  = (int*)(ws + OFF_SW);
    int* SB   = (int*)(ws + OFF_SB);
    float* WH = (float*)(ws + OFF_WH);
    float* WW = (float*)(ws + OFF_WW);
    float* WB = (float*)(ws + OFF_WB);
    float* B1 = (float*)(ws + OFF_B1);
    float* B2 = (float*)(ws + OFF_B2);
    float* out = (float*)d_out;

    params_kernel<<<1, 256, 0, stream>>>(P);
    build_weights<<<dim3(BATCH, NTILE), 64, 0, stream>>>(P, WH, WW, WB, SH, SWt, SB);

    // 1) cubic resize along H: x (256x256) -> B1 (224 rows, stride 256)
    gemm_rows<256, 65536ll, 256, 57344ll, 16>
        <<<dim3(BATCH * 3, NTILE, 2), 256, 0, stream>>>(x, B1, WH, SH);
    // 2) cubic resize along W: B1 -> B2 = xc (224x224)
    gemm_cols<256, 57344ll, 224, (long long)PIX, false>
        <<<dim3(BATCH * 3, NTILE, 2), 256, 0, stream>>>(B1, B2, WW, SWt, nullptr, P);
    // 3) per-image mean gray of clip(xc*bf)
    meangray_kernel<<<BATCH, 256, 0, stream>>>(B2, P);
    // 4) color jitter in place: B2 becomes xj
    jitter_kernel<<<(BATCH * PIX) / 256, 256, 0, stream>>>(B2, P);
    // 5) Gaussian blur along H: xj -> B1 (224x224, stride 224)
    gemm_rows<224, (long long)PIX, 224, (long long)PIX, 14>
        <<<dim3(BATCH * 3, NTILE, 2), 256, 0, stream>>>(B2, B1, WB, SB);
    // 6) Gaussian blur along W + where(apply_bl, xb, xj) -> d_out
    gemm_cols<224, (long long)PIX, 224, (long long)PIX, true>
        <<<dim3(BATCH * 3, NTILE, 2), 256, 0, stream>>>(B1, out, WB, SB, B2, P);
}